// VectorQuantizer_6038724018952
// MI455X (gfx1250) — compile-verified
//
#include <hip/hip_runtime.h>

// CDNA5 / gfx1250 vector-quantizer.
// argmin(||x-e||^2) recast as argmax(x.e_k - 0.5||e_k||^2) via fp32 WMMA
// (V_WMMA_F32_16X16X4_F32), -0.5||e||^2 folded into the C accumulator.
// x-tile staged into LDS via GLOBAL_LOAD_ASYNC_TO_LDS_B128 (ASYNCcnt).
// E pre-packed into WMMA B-fragment order so the K-loop feeds on single
// coalesced b64 loads from the L2-resident packed copy.
// Argmax = branchless sortable-key max (no exec-mask divergence).

typedef __attribute__((ext_vector_type(2))) float v2f;
typedef __attribute__((ext_vector_type(8))) float v8f;

#define NROWS 65536     // 16*64*64
#define DDIM  64
#define KCW   1024
#define ROWS_PER_WG 16
#define NWG (NROWS / ROWS_PER_WG)   // 4096

// ws layout (floats):
//   [0, 4096)        per-WG loss partials
//   [4096, 5120)     0.5*||e_k||^2
//   [5120, 70656)    E packed in B-fragment order (64 tiles * 16 steps * 32 lanes * v2f)
#define WS_ENORM 4096
#define WS_EPACK 5120
#define WS_NEED_PACKED ((WS_EPACK + 64 * 512 * 2) * sizeof(float))

// Monotone map IEEE float -> uint32 (order-preserving).
__device__ __forceinline__ unsigned f2ord(float f) {
    unsigned u = __float_as_uint(f);
    return (u & 0x80000000u) ? ~u : (u | 0x80000000u);
}
// Pack (score, col): 64-bit unsigned max == argmax, ties -> smaller col
// (matches jnp.argmin first-occurrence).
__device__ __forceinline__ unsigned long long packKey(float s, int col) {
    return ((unsigned long long)f2ord(s) << 32) |
           (unsigned long long)(unsigned)(KCW - 1 - col);
}
__device__ __forceinline__ int keyCol(unsigned long long k) {
    return (KCW - 1) - (int)(unsigned)(k & 0xFFFFFFFFull);
}

__global__ __launch_bounds__(256)
void vq_enorm_kernel(const float* __restrict__ E, float* __restrict__ ws) {
    int k = blockIdx.x * blockDim.x + threadIdx.x;   // 4 blocks x 256 = 1024
    float s = 0.0f;
#pragma unroll
    for (int d = 0; d < DDIM; ++d) {
        float e = E[(size_t)d * KCW + k];
        s += e * e;
    }
    ws[WS_ENORM + k] = 0.5f * s;
}

// Re-lay E into per-(tile, k-step, lane) v2f fragments: one b64 per lane per
// WMMA step in the main loop. t = ((kt*16)+s)*32 + lane, 32768 threads total.
__global__ __launch_bounds__(256)
void vq_pack_kernel(const float* __restrict__ E, float* __restrict__ ws) {
    int t    = blockIdx.x * blockDim.x + threadIdx.x;  // 128 blocks x 256
    int kt   = t >> 9;          // column tile (16 codewords)
    int rem  = t & 511;
    int s    = rem >> 5;        // k-step (4 of D per WMMA)
    int l    = rem & 31;
    int half = l >> 4;
    int lm   = l & 15;
    int col  = kt * 16 + lm;
    int k0   = 4 * s + 2 * half;
    float* dst = ws + WS_EPACK + (size_t)t * 2;
    dst[0] = E[(size_t)k0 * KCW + col];
    dst[1] = E[(size_t)(k0 + 1) * KCW + col];
}

template <bool PACKED>
__global__ __launch_bounds__(256)
void vq_main_kernel(const float* __restrict__ X, const float* __restrict__ E,
                    float* __restrict__ outQ, float* __restrict__ outIdx,
                    float* __restrict__ ws) {
    __shared__ float xs[ROWS_PER_WG * DDIM];            // 4 KB x-tile
    __shared__ unsigned long long redK[8][16];          // per-wave winners
    __shared__ int   bestColSh[16];
    __shared__ float lred[256];

    const int tid  = threadIdx.x;
    const int wave = tid >> 5;
    const int lane = tid & 31;
    const int half = lane >> 4;     // which 16-lane half of the wave
    const int lm   = lane & 15;
    const size_t rowBase = (size_t)blockIdx.x * ROWS_PER_WG;
    const size_t xBase   = rowBase * DDIM;     // contiguous 1024 floats

    // --- Async-stage the 16x64 x-tile into LDS (16 B per thread). -----------
    // Generic LDS addresses are {SHARED_BASE, lds_off[31:0]}; the low 32 bits
    // of the shared pointer are the LDS byte address the VDST operand wants.
    {
        unsigned lds_off =
            (unsigned)(unsigned long long)(uintptr_t)(xs + (size_t)tid * 4);
        unsigned long long gaddr =
            (unsigned long long)(uintptr_t)(X + xBase + (size_t)tid * 4);
        asm volatile("global_load_async_to_lds_b128 %0, %1, off"
                     :: "v"(lds_off), "v"(gaddr) : "memory");
        asm volatile("s_wait_asynccnt 0x0" ::: "memory");
    }
    __syncthreads();

    // A fragments (16x4 fp32 per WMMA): lanes 0-15 -> K=k0,k0+1 of row lm,
    // lanes 16-31 -> K=k0+2,k0+3. Built once, reused for all 8 column tiles.
    v2f aFrag[16];
#pragma unroll
    for (int s = 0; s < 16; ++s) {
        int k0 = 4 * s + 2 * half;
        v2f a;
        a.x = xs[lm * DDIM + k0];
        a.y = xs[lm * DDIM + k0 + 1];
        aFrag[s] = a;
    }

    const float* __restrict__ enormhalf = ws + WS_ENORM;
    const v2f*   __restrict__ Ep = (const v2f*)(ws + WS_EPACK);

    // Per-lane running best over this wave's 8 column tiles (branchless).
    unsigned long long bestK[8];
#pragma unroll
    for (int r = 0; r < 8; ++r) bestK[r] = 0ull;

    for (int j = 0; j < 8; ++j) {
        const int kt = wave * 8 + j;       // column tile index
        const int c0 = kt * 16;
        const float negE = -enormhalf[c0 + lm];
        v8f acc;
#pragma unroll
        for (int r = 0; r < 8; ++r) acc[r] = negE;   // C init = -0.5||e||^2

        if constexpr (PACKED) {
            const v2f* __restrict__ Ept = Ep + (size_t)kt * 512 + lane;
#pragma unroll
            for (int s = 0; s < 16; ++s) {           // one b64 per step
                v2f b = Ept[s * 32];
                acc = __builtin_amdgcn_wmma_f32_16x16x4_f32(
                    false, aFrag[s], false, b, (short)0, acc, false, false);
            }
        } else {
#pragma unroll
            for (int s = 0; s < 16; ++s) {           // two b32 per step
                int k0 = 4 * s + 2 * half;
                v2f b;
                b.x = E[(size_t)k0 * KCW + c0 + lm];
                b.y = E[(size_t)(k0 + 1) * KCW + c0 + lm];
                acc = __builtin_amdgcn_wmma_f32_16x16x4_f32(
                    false, aFrag[s], false, b, (short)0, acc, false, false);
            }
        }

#pragma unroll
        for (int r = 0; r < 8; ++r) {
            unsigned long long k = packKey(acc[r], c0 + lm);
            bestK[r] = (k > bestK[r]) ? k : bestK[r];
        }
    }

    // Final xor-shuffle max within each 16-lane half (row r+8*half's columns
    // live across the half). Branchless v_cndmask selects.
#pragma unroll
    for (int r = 0; r < 8; ++r) {
        unsigned long long k = bestK[r];
#pragma unroll
        for (int off = 8; off >= 1; off >>= 1) {
            unsigned long long k2 = __shfl_xor(k, off, 32);
            k = (k2 > k) ? k2 : k;
        }
        bestK[r] = k;
    }

    // Per-wave winners to LDS: half 0 covers rows 0-7, half 1 rows 8-15.
    if (lm == 0) {
#pragma unroll
        for (int r = 0; r < 8; ++r) redK[wave][r + 8 * half] = bestK[r];
    }
    __syncthreads();

    // Combine across the 8 waves; emit indices.
    if (tid < 16) {
        unsigned long long k = redK[0][tid];
#pragma unroll
        for (int w = 1; w < 8; ++w) {
            unsigned long long k2 = redK[w][tid];
            k = (k2 > k) ? k2 : k;
        }
        int c = keyCol(k);
        bestColSh[tid] = c;
        outIdx[rowBase + tid] = (float)c;
    }
    __syncthreads();

    // Gather quantized vectors (straight-through output == codeword) and
    // accumulate loss partial sum((q-x)^2).
    float lsum = 0.0f;
#pragma unroll
    for (int i = 0; i < 4; ++i) {
        int e   = tid * 4 + i;
        int row = e >> 6;
        int d   = e & 63;
        float q = E[(size_t)d * KCW + bestColSh[row]];
        outQ[xBase + e] = q;
        float diff = q - xs[e];
        lsum += diff * diff;
    }
    lred[tid] = lsum;
    __syncthreads();
#pragma unroll
    for (int s = 128; s > 0; s >>= 1) {
        if (tid < s) lred[tid] += lred[tid + s];
        __syncthreads();
    }
    if (tid == 0) ws[blockIdx.x] = lred[0];
}

__global__ __launch_bounds__(256)
void vq_finalize_kernel(const float* __restrict__ ws, float* __restrict__ outLoss) {
    __shared__ float lred[256];
    int tid = threadIdx.x;
    float s = 0.0f;
#pragma unroll
    for (int i = 0; i < 16; ++i) s += ws[tid * 16 + i];   // fixed order: deterministic
    lred[tid] = s;
    __syncthreads();
#pragma unroll
    for (int k = 128; k > 0; k >>= 1) {
        if (tid < k) lred[tid] += lred[tid + k];
        __syncthreads();
    }
    // loss = q_latent + 0.25*e_latent = 1.25 * mean((q-x)^2)
    if (tid == 0) outLoss[0] = 1.25f * lred[0] / (float)(NROWS * DDIM);
}

extern "C" void kernel_launch(void* const* d_in, const int* in_sizes, int n_in,
                              void* d_out, int out_size, void* d_ws, size_t ws_size,
                              hipStream_t stream) {
    const float* X = (const float*)d_in[0];   // (16,64,64,64) f32
    const float* E = (const float*)d_in[1];   // (64,1024) f32
    float* out     = (float*)d_out;
    float* ws      = (float*)d_ws;

    float* outQ    = out;                          // 4,194,304 floats
    float* outIdx  = out + (size_t)NROWS * DDIM;   // 65,536 indices (as float)
    float* outLoss = outIdx + NROWS;               // 1 scalar

    vq_enorm_kernel<<<KCW / 256, 256, 0, stream>>>(E, ws);
    if (ws_size >= WS_NEED_PACKED) {
        vq_pack_kernel<<<128, 256, 0, stream>>>(E, ws);
        vq_main_kernel<true><<<NWG, 256, 0, stream>>>(X, E, outQ, outIdx, ws);
    } else {
        vq_main_kernel<false><<<NWG, 256, 0, stream>>>(X, E, outQ, outIdx, ws);
    }
    vq_finalize_kernel<<<1, 256, 0, stream>>>(ws, outLoss);
}